// ChebyshevAdditiveAngularMargin_970662609120
// MI455X (gfx1250) — compile-verified
//
#include <hip/hip_runtime.h>

// ---------------------------------------------------------------------------
// ChebyshevAdditiveAngularMargin for MI455X (gfx1250, wave32)
//
// out = SCALE * ( targets*phi + (1-targets)*cos )  with phi = Cheb30(cos)
// Pure streaming op: 768 MB traffic, ~33 us floor at 23.3 TB/s.
// targets is one-hot -> skip the Clenshaw recurrence with a wave32 ballot
// unless some lane in the wave actually has a nonzero target.
// ---------------------------------------------------------------------------

typedef float        fv4 __attribute__((ext_vector_type(4)));
typedef unsigned int uv4 __attribute__((ext_vector_type(4)));
typedef int          iv4 __attribute__((ext_vector_type(4)));
typedef int          iv8 __attribute__((ext_vector_type(8)));

#define NCOEF     31
#define ARC_SCALE 30.0f
#define ARC_TH    (-0.98006657784124163f)   /* cos(pi - 0.2)        */
#define ARC_MM    (0.039733866159012245f)   /* sin(pi - 0.2) * 0.2  */
#define CLIP_LO   (-1.0f + 1.0e-7f)
#define CLIP_HI   (1.0f - 1.0e-7f)

#ifndef __has_builtin
#define __has_builtin(x) 0
#endif

// TDM (Tensor Data Mover) staging of coefficients into LDS.
#if defined(__HIP_DEVICE_COMPILE__) &&                      \
    __has_builtin(__builtin_amdgcn_tensor_load_to_lds) &&   \
    __has_builtin(__builtin_amdgcn_s_wait_tensorcnt)
#define USE_TDM 1
#else
#define USE_TDM 0
#endif

// amdgpu-toolchain (therock headers) exposes the 6-arg builtin; ROCm 7.2 the 5-arg.
#if defined(__has_include)
#if __has_include(<hip/amd_detail/amd_gfx1250_TDM.h>)
#define TDM_SIX_ARG 1
#else
#define TDM_SIX_ARG 0
#endif
#else
#define TDM_SIX_ARG 0
#endif

#if defined(__HIP_DEVICE_COMPILE__)
#define WAVE_ANY(p) (__builtin_amdgcn_ballot_w32(p) != 0u)
#else
#define WAVE_ANY(p) (p)
#endif

// Clenshaw: b_k = c_k + 2x*b_{k+1} - b_{k+2}; f = b0 - x*b1. Coeffs from LDS.
__device__ __forceinline__ float cheb_phi(float c, const float* sc) {
  const float x2 = c + c;
  float b1 = 0.0f, b2 = 0.0f;
#pragma unroll
  for (int k = NCOEF - 1; k >= 0; --k) {
    const float b = __builtin_fmaf(x2, b1, sc[k] - b2);
    b2 = b1;
    b1 = b;
  }
  const float ph = __builtin_fmaf(-c, b2, b1);  // b0 - x*b1
  return (c > ARC_TH) ? ph : (c - ARC_MM);
}

__global__ __launch_bounds__(256) void cheb_margin_kernel(
    const fv4* __restrict__ out4, const fv4* __restrict__ tgt4,
    const float* __restrict__ coeffs, fv4* __restrict__ res4, long n4) {
  __shared__ float s_coef[32];

#if USE_TDM
  // One TDM DMA per workgroup: 31 fp32 coeffs -> LDS. Issued by wave 0 only
  // (TENSORcnt is per-wave, so the issuing wave waits, then the workgroup
  // barrier publishes the LDS data to the other waves).
  if (threadIdx.x < 32u) {
    const unsigned long long ga = (unsigned long long)(const void*)coeffs;
    const unsigned int lds_addr = (unsigned int)(unsigned long long)(void*)&s_coef[0];
    // D# group 0: count=1 | lds_addr | global_addr[56:0] | type=2
    uv4 g0;
    g0.x = 1u;
    g0.y = lds_addr;
    g0.z = (unsigned int)(ga & 0xFFFFFFFFull);
    g0.w = ((unsigned int)(ga >> 32) & 0x01FFFFFFu) | (2u << 30);
    // D# group 1: data_size=4B, tensor_dim0=31, tensor_dim1=1,
    //             tile_dim0=32 (last element OOB -> reads as 0), tile_dim1=1,
    //             tensor_dim0_stride=32.
    iv8 g1;
    g1[0] = 0x00020000;          // workgroup_mask=0, data_size=2 (4 bytes)
    g1[1] = (int)(31u << 16);    // tensor_dim0[15:0]
    g1[2] = (int)(1u << 16);     // tensor_dim1[15:0]
    g1[3] = (int)(32u << 16);    // tile_dim0
    g1[4] = 1;                   // tile_dim1 = 1, tile_dim2 = 0
    g1[5] = 32;                  // tensor_dim0_stride (low 32)
    g1[6] = 0;
    g1[7] = 0;
    const iv4 z4 = {0, 0, 0, 0};
#if TDM_SIX_ARG
    const iv8 z8 = {0, 0, 0, 0, 0, 0, 0, 0};
    __builtin_amdgcn_tensor_load_to_lds(g0, g1, z4, z4, z8, 0);
#else
    __builtin_amdgcn_tensor_load_to_lds(g0, g1, z4, z4, 0);
#endif
    __builtin_amdgcn_s_wait_tensorcnt(0);
  }
#else
  if (threadIdx.x < (unsigned)NCOEF) s_coef[threadIdx.x] = coeffs[threadIdx.x];
#endif
  __syncthreads();

  const unsigned tid = threadIdx.x;
  const long stride = (long)blockDim.x;
  const long base = (long)blockIdx.x * (stride * 4) + (long)tid;

  // Issue all 8 B128 non-temporal loads up front (no reuse -> bypass caches).
  fv4 o[4], t[4];
#pragma unroll
  for (int u = 0; u < 4; ++u) {
    const long i = base + (long)u * stride;
    if (i < n4) {
      o[u] = __builtin_nontemporal_load(&out4[i]);
      t[u] = __builtin_nontemporal_load(&tgt4[i]);
    }
  }

#pragma unroll
  for (int u = 0; u < 4; ++u) {
    const long i = base + (long)u * stride;
    if (i >= n4) continue;

    fv4 c;
    c.x = fminf(fmaxf(o[u].x, CLIP_LO), CLIP_HI);
    c.y = fminf(fmaxf(o[u].y, CLIP_LO), CLIP_HI);
    c.z = fminf(fmaxf(o[u].z, CLIP_LO), CLIP_HI);
    c.w = fminf(fmaxf(o[u].w, CLIP_LO), CLIP_HI);

    // Fast path: no target in this wave's slice -> out = SCALE * cosine.
    fv4 r = ARC_SCALE * c;

    const bool hit = (t[u].x != 0.0f) | (t[u].y != 0.0f) |
                     (t[u].z != 0.0f) | (t[u].w != 0.0f);
    if (WAVE_ANY(hit)) {
      // Rare path (~6% of waves for one-hot targets): full Clenshaw + blend.
      r.x = ARC_SCALE * __builtin_fmaf(t[u].x, cheb_phi(c.x, s_coef) - c.x, c.x);
      r.y = ARC_SCALE * __builtin_fmaf(t[u].y, cheb_phi(c.y, s_coef) - c.y, c.y);
      r.z = ARC_SCALE * __builtin_fmaf(t[u].z, cheb_phi(c.z, s_coef) - c.z, c.z);
      r.w = ARC_SCALE * __builtin_fmaf(t[u].w, cheb_phi(c.w, s_coef) - c.w, c.w);
    }
    __builtin_nontemporal_store(r, &res4[i]);
  }
}

extern "C" void kernel_launch(void* const* d_in, const int* in_sizes, int n_in,
                              void* d_out, int out_size, void* d_ws, size_t ws_size,
                              hipStream_t stream) {
  const float* outputs = (const float*)d_in[0];  // (N, C) fp32
  const float* targets = (const float*)d_in[1];  // (N, C) fp32 one-hot
  const float* coeffs  = (const float*)d_in[2];  // (31,) fp32
  float* res = (float*)d_out;

  const long n  = (long)in_sizes[0];  // 8192*8192, divisible by 4
  const long n4 = n >> 2;

  const int  threads   = 256;
  const long per_block = (long)threads * 4;  // 4 float4 per thread
  const int  blocks    = (int)((n4 + per_block - 1) / per_block);

  cheb_margin_kernel<<<dim3(blocks), dim3(threads), 0, stream>>>(
      (const fv4*)outputs, (const fv4*)targets, coeffs, (fv4*)res, n4);
}